// MusicEventEmbedding_34926674051700
// MI455X (gfx1250) — compile-verified
//
#include <hip/hip_runtime.h>
#include <hip/hip_bf16.h>
#include <stdint.h>

// MusicEventEmbedding for MI455X (gfx1250).
// BATCH=8, SEQ=8192, DIM=512, VOCAB=44.
// Kernel 1: parallel segment-id scan (1 block per batch row).
// Kernel 2: TDM-staged embedding table in LDS + on-the-fly positional encoding.

#define SEQ 8192
#define DIM 512
#define VOCAB 44
#define SCAN_THREADS 1024
#define CHUNK (SEQ / SCAN_THREADS)   // 8 tokens per thread

typedef unsigned int u32x4 __attribute__((ext_vector_type(4)));
typedef int          i32x8 __attribute__((ext_vector_type(8)));
typedef int          i32x4 __attribute__((ext_vector_type(4)));

// ---------------------------------------------------------------------------
// Kernel 1: segment ids.
// seg[t] = cumsum(flags[0..t]),  flags[0]=0,
// flags[t] (t>=1): c=tok[t], p=tok[t-1]
//   note (c<12):   flag = (p>=12) || nc_pos
//   else:          flag = (p<12)
// nc_pos  <=>  exists reset (tok in [36,41]) at L<=t-1 with (#notes in (L,t-1]) <= 1.
// Parallel form: N = prefix note-count, R = prefix-max reset position.
// ---------------------------------------------------------------------------
__global__ __launch_bounds__(SCAN_THREADS)
void seg_scan_kernel(const int* __restrict__ tokens, int* __restrict__ seg) {
    extern __shared__ int lds[];
    int* Narr = lds;               // 8192 ints
    int* Rarr = lds + SEQ;         // 8192 ints
    int* sA   = lds + 2 * SEQ;     // 1024 ints
    int* sB   = sA + SCAN_THREADS; // 1024 ints

    const int tid  = threadIdx.x;
    const int row  = blockIdx.x;
    const int base = tid * CHUNK;
    const int gbase = row * SEQ + base;

    int toks[CHUNK];
#pragma unroll
    for (int k = 0; k < CHUNK; ++k) toks[k] = tokens[gbase + k];

    // ---- phase 1: local (note-count, last-reset) ----
    int ln = 0, lr = 0;
    int locN[CHUNK], locR[CHUNK];
#pragma unroll
    for (int k = 0; k < CHUNK; ++k) {
        int i = base + k;
        if (i >= 1) {
            int c = toks[k];
            if (c < 12) ln++;
            if (c >= 36 && c <= 41) lr = i;   // positions >=1, 0 == "none"
        }
        locN[k] = ln;
        locR[k] = lr;
    }
    sA[tid] = ln; sB[tid] = lr;
    __syncthreads();
    // inclusive Hillis-Steele scan of (sum, max) over 1024 partials
    for (int off = 1; off < SCAN_THREADS; off <<= 1) {
        int a = sA[tid], b = sB[tid];
        int a2 = 0, b2 = 0;
        if (tid >= off) { a2 = sA[tid - off]; b2 = sB[tid - off]; }
        __syncthreads();
        sA[tid] = a + a2;
        sB[tid] = (b > b2) ? b : b2;
        __syncthreads();
    }
    int exN = 0, exR = 0;
    if (tid > 0) { exN = sA[tid - 1]; exR = sB[tid - 1]; }
#pragma unroll
    for (int k = 0; k < CHUNK; ++k) {
        int r = exR > locR[k] ? exR : locR[k];
        Narr[base + k] = exN + locN[k];
        Rarr[base + k] = r;
    }
    __syncthreads();

    // ---- phase 2: flags + cumsum -> seg ----
    int prevTok = (base == 0) ? 0 : tokens[gbase - 1];
    int lf = 0;
    int locF[CHUNK];
#pragma unroll
    for (int k = 0; k < CHUNK; ++k) {
        int i = base + k;
        int f = 0;
        if (i >= 1) {
            int c = toks[k];
            int p = (k == 0) ? prevTok : toks[k - 1];
            if (c < 12) {
                int L = Rarr[i - 1];
                int ncpos = 0;
                if (L > 0) ncpos = (Narr[i - 1] - Narr[L]) <= 1;
                f = (p >= 12) || ncpos;
            } else {
                f = (p < 12);
            }
        }
        lf += f;
        locF[k] = lf;
    }
    sA[tid] = lf;
    __syncthreads();
    for (int off = 1; off < SCAN_THREADS; off <<= 1) {
        int a = sA[tid];
        int a2 = 0;
        if (tid >= off) a2 = sA[tid - off];
        __syncthreads();
        sA[tid] = a + a2;
        __syncthreads();
    }
    int exF = (tid > 0) ? sA[tid - 1] : 0;
#pragma unroll
    for (int k = 0; k < CHUNK; ++k)
        seg[gbase + k] = exF + locF[k];
}

// ---------------------------------------------------------------------------
// Kernel 2: out[t, d] = tableLDS[tok[t]][d] * sqrt(512) + PE(seg[t], d)
// 256 threads = 8 wave32s; 1 token per wave per step; lane writes 4x float4.
// Table staged into LDS once per block via the Tensor Data Mover.
// ---------------------------------------------------------------------------
#define TOK_PER_BLOCK 128

__global__ __launch_bounds__(256)
void emb_pe_kernel(const int* __restrict__ tokens,
                   const float* __restrict__ table,
                   const int* __restrict__ seg,
                   float* __restrict__ out) {
    extern __shared__ float tbl[];   // VOCAB*DIM floats = 90112 bytes

    // --- stage embedding table into LDS with the TDM (wave 0 only) ---
    if (threadIdx.x < 32) {
        unsigned long long ga = (unsigned long long)(uintptr_t)table;
        unsigned ldsa = (unsigned)(uintptr_t)(void*)tbl;  // low 32 bits = LDS offset
        u32x4 g0;
        g0[0] = 1u;                                       // count=1, is_restore=0
        g0[1] = ldsa;                                     // lds_addr
        g0[2] = (unsigned)(ga & 0xffffffffu);             // global_addr[31:0]
        g0[3] = (unsigned)((ga >> 32) & 0x1ffffffu) | (2u << 30); // addr[56:32] | type=2
        i32x8 g1;
        g1[0] = (int)0x00020000u;        // workgroup_mask=0, data_size=2 (4 bytes)
        g1[1] = (int)(512u << 16);       // tensor_dim0[15:0] = 512 (bits 79:48)
        g1[2] = (int)(44u << 16);        // tensor_dim0 hi=0 | tensor_dim1[15:0]=44
        g1[3] = (int)(512u << 16);       // tensor_dim1 hi=0 | tile_dim0=512
        g1[4] = 44;                      // tile_dim1=44 | tile_dim2=0 (2-D)
        g1[5] = 512;                     // tensor_dim0_stride low 32
        g1[6] = (int)(22528u << 16);     // stride0 hi16=0 | tensor_dim1_stride lo16
        g1[7] = 0;                       // tensor_dim1_stride hi
        i32x4 g2 = {0, 0, 0, 0};
        i32x4 g3 = {0, 0, 0, 0};
        i32x8 g4 = {0, 0, 0, 0, 0, 0, 0, 0};  // unused trailing group (6-arg builtin)
        __builtin_amdgcn_tensor_load_to_lds(g0, g1, g2, g3, g4, 0);
        __builtin_amdgcn_s_wait_tensorcnt(0);
    }
    __syncthreads();

    const int lane = threadIdx.x & 31;
    const int wave = threadIdx.x >> 5;

    // PE frequencies for this lane's 8 (k) slots: d = q*128 + lane*4 .. +3
    const float C2 = -13.287712379549449f / 256.0f;   // -log2(10000)/256
    float divv[8];
#pragma unroll
    for (int q = 0; q < 4; ++q) {
        int k0 = (q * 128 + lane * 4) >> 1;
        divv[2 * q]     = exp2f((float)k0 * C2);
        divv[2 * q + 1] = exp2f((float)(k0 + 1) * C2);
    }
    const float scale = 22.627416997969522f;  // sqrt(512)

    const int blockStart = blockIdx.x * TOK_PER_BLOCK;
    for (int j = wave; j < TOK_PER_BLOCK; j += 8) {
        int t = blockStart + j;
        int tk = tokens[t];
        float s = (float)seg[t];
        const float* rowp = tbl + tk * DIM;
        size_t obase = (size_t)t * DIM;
#pragma unroll
        for (int q = 0; q < 4; ++q) {
            int d = q * 128 + lane * 4;
            float4 e = *(const float4*)(rowp + d);   // ds_load_b128, bank-balanced
            float a0 = s * divv[2 * q];
            float a1 = s * divv[2 * q + 1];
            float4 r;
            r.x = fmaf(e.x, scale, __sinf(a0));
            r.y = fmaf(e.y, scale, __cosf(a0));
            r.z = fmaf(e.z, scale, __sinf(a1));
            r.w = fmaf(e.w, scale, __cosf(a1));
            *(float4*)(out + obase + d) = r;         // 512B coalesced per wave
        }
    }
}

extern "C" void kernel_launch(void* const* d_in, const int* in_sizes, int n_in,
                              void* d_out, int out_size, void* d_ws, size_t ws_size,
                              hipStream_t stream) {
    const int*   tokens = (const int*)d_in[0];     // harness: integer -> const int*
    const float* table  = (const float*)d_in[1];   // (44, 512) fp32
    float*       out    = (float*)d_out;           // (8, 8192, 512) fp32
    int*         seg    = (int*)d_ws;              // 8*8192 ints = 256 KB scratch

    // 1 block per batch row; LDS: N(32K) + R(32K) + scanA(4K) + scanB(4K)
    size_t scanLds = (size_t)(2 * SEQ + 2 * SCAN_THREADS) * sizeof(int);
    seg_scan_kernel<<<8, SCAN_THREADS, scanLds, stream>>>(tokens, seg);

    // 65536 tokens / 128 per block = 512 blocks; LDS: 44*512 floats
    size_t tblLds = (size_t)VOCAB * DIM * sizeof(float);
    emb_pe_kernel<<<(8 * SEQ) / TOK_PER_BLOCK, 256, tblLds, stream>>>(tokens, table, seg, out);
}